// CreateGraphCondensation_6906307412022
// MI455X (gfx1250) — compile-verified
//
#include <hip/hip_runtime.h>
#include <hip/hip_bf16.h>

// ---------------------------------------------------------------------------
// CreateGraphCondensation for gfx1250 (MI455X)
//   out = (rs_down[3], rs_up[3], nidx_down[N*64], distsq_down[N*64],
//          sel_idx_up[M], weights_down[N*64])  -- all as float in d_out
// Fused tiled distance computation with V_WMMA_F32_16X16X4_F32
// (A row = (-2x,-2y,-2z,1), B col = (x,y,z,|x|^2)), 2 tiles / WMMAs per
// wave-iteration, register-resident row metadata (bitmask validity),
// per-lane sorted top lists + wave-wide argmin merge.
// ---------------------------------------------------------------------------

#define THRV   0.5f
#define KNN    64
#define TS     16
#define NWAVE  8
#define TPI    2                  // column tiles (and WMMAs) per iteration
#define BLK    (NWAVE * 32)
#define LLEN   16                 // per-lane local sorted top list
#define CPR    (NWAVE * 2 * LLEN) // merge candidates per row = 256
#define FINF   3.0e38f

typedef float v2f __attribute__((ext_vector_type(2)));
typedef float v8f __attribute__((ext_vector_type(8)));

__global__ __launch_bounds__(BLK)
void knn_down_kernel(const float* __restrict__ score,
                     const float* __restrict__ coords,
                     const int*   __restrict__ rs,
                     float* __restrict__ out_nidx,
                     float* __restrict__ out_dist,
                     float* __restrict__ out_w,
                     int N)
{
    __shared__ float s_tile[NWAVE][TPI * TS * TS];
    __shared__ float s_rowsq[TS];
    __shared__ int   s_rowseg[TS];   // 0/1 (two segments: rs = [0, s, N])
    __shared__ int   s_rowprom[TS];
    __shared__ float s_cd[TS][CPR];
    __shared__ int   s_ci[TS][CPR];
    __shared__ float s_seld[TS][KNN + 1];
    __shared__ int   s_seli[TS][KNN + 1];
    __shared__ float s_sum[TS];

    const int tid  = threadIdx.x;
    const int lane = tid & 31;
    const int w    = tid >> 5;
    const int i0   = blockIdx.x * TS;
    const int rs1  = rs[1];

    // ---- row metadata (shared by all waves in the block) ----
    if (tid < TS) {
        int gi = i0 + tid;
        float x = coords[3 * gi + 0];
        float y = coords[3 * gi + 1];
        float z = coords[3 * gi + 2];
        s_rowsq[tid]   = x * x + y * y + z * z;
        s_rowseg[tid]  = (gi >= rs1) ? 1 : 0;
        s_rowprom[tid] = (score[gi] > THRV) ? 1 : 0;
    }
    __syncthreads();

    // ---- hoist row metadata into registers (loop-invariant) ----
    const int mbase = (lane < 16) ? 0 : 8;   // D rows this lane owns: mbase..mbase+7
    float rsq[8];
#pragma unroll
    for (int r = 0; r < 8; ++r) rsq[r] = s_rowsq[mbase + r];

    unsigned segmask = 0u, prommask = 0u;
#pragma unroll
    for (int m = 0; m < TS; ++m) {
        segmask  |= ((unsigned)s_rowseg[m])  << m;
        prommask |= ((unsigned)s_rowprom[m]) << m;
    }

    // ---- A operand (16x4 f32, 2 VGPRs/lane): row = (-2x,-2y,-2z,1) ----
    const int ra = i0 + (lane & 15);
    const float ax = coords[3 * ra + 0];
    const float ay = coords[3 * ra + 1];
    const float az = coords[3 * ra + 2];
    v2f A;
    A.x = (lane < 16) ? (-2.0f * ax) : (-2.0f * az);
    A.y = (lane < 16) ? (-2.0f * ay) : 1.0f;

    // ---- per-lane sorted top list (ascending) ----
    float ld[LLEN];
    int   li[LLEN];
#pragma unroll
    for (int s = 0; s < LLEN; ++s) { ld[s] = FINF; li[s] = -1; }

    const int ntiles = N / TS;           // 512
    const int nn     = lane & 15;

    for (int t = w * TPI; t < ntiles; t += NWAVE * TPI) {
        // prefetch this wave's next tile pair
        const int tn = t + NWAVE * TPI;
        if (tn < ntiles)
            __builtin_prefetch(&coords[3 * (tn * TS + nn)], 0, 3);

        v8f      C[TPI];
        unsigned vm[TPI];
        int      cb[TPI];

#pragma unroll
        for (int u = 0; u < TPI; ++u) {
            const int j0 = (t + u) * TS;
            cb[u] = j0 + nn;
            const float bx  = coords[3 * cb[u] + 0];
            const float by  = coords[3 * cb[u] + 1];
            const float bz  = coords[3 * cb[u] + 2];
            const float bsq = bx * bx + by * by + bz * bz;
            const int cseg  = (cb[u] >= rs1) ? 1 : 0;
            const int cprom = (score[cb[u]] > THRV) ? 1 : 0;

            // per-column validity bitmask over the 16 rows (pure VALU later)
            unsigned m = (cseg ? segmask : (~segmask & 0xFFFFu));
            m &= (~prommask & 0xFFFFu);
            vm[u] = cprom ? m : 0u;

            // B operand (4x16 f32): col = (x, y, z, |x|^2)
            v2f B;
            B.x = (lane < 16) ? bx : bz;
            B.y = (lane < 16) ? by : bsq;

            v8f Z = {};
            // D[i][j] = |x_j|^2 - 2 x_i . x_j   (add |x_i|^2 below)
            C[u] = __builtin_amdgcn_wmma_f32_16x16x4_f32(
                       false, A, false, B, (short)0, Z, false, false);
        }

        // ---- mask + stage tiles into this wave's LDS buffer ----
#pragma unroll
        for (int u = 0; u < TPI; ++u) {
#pragma unroll
            for (int r = 0; r < 8; ++r) {
                const int M = mbase + r;
                float d2 = fmaxf(C[u][r] + rsq[r], 0.0f);
                const bool valid = (vm[u] >> M) & 1u;
                const bool self  = (i0 + M) == cb[u];
                float dm = self ? -1.0f : (valid ? d2 : FINF);
                s_tile[w][u * (TS * TS) + M * TS + nn] = dm;
            }
        }
        __syncthreads();   // uniform: every wave runs ntiles/(NWAVE*TPI) iters

        // ---- scan tiles (lane -> row view) and insert into top list ----
        const int row = lane & 15;
        const int cof = (lane >> 4) * 8;
#pragma unroll
        for (int u = 0; u < TPI; ++u) {
#pragma unroll
            for (int c = 0; c < 8; ++c) {
                const float dm  = s_tile[w][u * (TS * TS) + row * TS + cof + c];
                const int   idx = (t + u) * TS + cof + c;
                if (dm < ld[LLEN - 1]) {
                    ld[LLEN - 1] = dm;
                    li[LLEN - 1] = idx;
#pragma unroll
                    for (int s = LLEN - 1; s > 0; --s) {
                        if (ld[s] < ld[s - 1]) {
                            float td = ld[s]; ld[s] = ld[s - 1]; ld[s - 1] = td;
                            int   ti = li[s]; li[s] = li[s - 1]; li[s - 1] = ti;
                        }
                    }
                }
            }
        }
    }

    // ---- dump per-lane lists into per-row merge arrays ----
    {
        const int row  = lane & 15;
        const int base = w * (2 * LLEN) + (lane >> 4) * LLEN;
#pragma unroll
        for (int s = 0; s < LLEN; ++s) {
            s_cd[row][base + s] = ld[s];
            s_ci[row][base + s] = li[s];
        }
    }
    __syncthreads();

    // ---- per-row selection of K+1 smallest (wave-wide argmin) ----
    for (int rr = 0; rr < 2; ++rr) {
        const int row = w + rr * NWAVE;   // 0..15
        float e_d[8];
        int   e_i[8];
#pragma unroll
        for (int k = 0; k < 8; ++k) {
            e_d[k] = s_cd[row][lane * 8 + k];
            e_i[k] = s_ci[row][lane * 8 + k];
        }
        for (int s = 0; s < KNN + 1; ++s) {
            float bd = FINF;
            int   bc = 0x7FFFFFFF;
#pragma unroll
            for (int k = 0; k < 8; ++k)
                if (e_d[k] < bd) { bd = e_d[k]; bc = (lane << 3) | k; }
#pragma unroll
            for (int off = 16; off > 0; off >>= 1) {
                float od = __shfl_xor(bd, off, 32);
                int   oc = __shfl_xor(bc, off, 32);
                if (od < bd || (od == bd && oc < bc)) { bd = od; bc = oc; }
            }
            if (lane == 0) { s_seld[row][s] = FINF; s_seli[row][s] = -1; }
            if (bc != 0x7FFFFFFF && (bc >> 3) == lane) {
                const int k = bc & 7;
                int widx = -1;
#pragma unroll
                for (int kk = 0; kk < 8; ++kk) if (kk == k) widx = e_i[kk];
                s_seld[row][s] = bd;
                s_seli[row][s] = widx;
#pragma unroll
                for (int kk = 0; kk < 8; ++kk) if (kk == k) e_d[kk] = FINF;
            }
        }
    }
    __syncthreads();

    // ---- per-row softmax(exp(-d2)) denominator over slots 1..K ----
    for (int rr = 0; rr < 2; ++rr) {
        const int row = w + rr * NWAVE;
        float part = 0.0f;
        for (int s = 1 + lane; s <= KNN; s += 32) {
            float d  = s_seld[row][s];
            float de = (d >= 1.0e37f) ? 0.0f : d;
            part += expf(expf(-de));
        }
#pragma unroll
        for (int off = 16; off > 0; off >>= 1)
            part += __shfl_xor(part, off, 32);
        if (lane == 0) s_sum[row] = part;
    }
    __syncthreads();

    // ---- write outputs: nidx, distsq, weights (skip slot 0 = self) ----
    for (int q = tid; q < TS * KNN; q += BLK) {
        const int row = q >> 6;
        const int s   = (q & 63) + 1;
        const float d  = s_seld[row][s];
        const int   ix = s_seli[row][s];
        const bool  ok = d < 1.0e37f;
        const float de = ok ? fmaxf(d, 0.0f) : 0.0f;
        const long  o  = (long)(i0 + row) * KNN + (s - 1);
        out_nidx[o] = (float)(ok ? ix : -1);
        out_dist[o] = de;
        out_w[o]    = expf(expf(-de)) / s_sum[row];
    }
}

// ---------------------------------------------------------------------------
// Small kernel: rs_down copy, rs_up (promoted prefix counts), sel_idx_up
// via single-block prefix scan.
// ---------------------------------------------------------------------------
__global__ __launch_bounds__(1024)
void compact_kernel(const float* __restrict__ score,
                    const int*   __restrict__ rs,
                    float* __restrict__ out,
                    int N, int off_sel, int M)
{
    __shared__ int s_scan[1024];
    __shared__ int s_c1, s_c2;

    const int tid = threadIdx.x;
    if (tid == 0) { s_c1 = 0; s_c2 = 0; }
    __syncthreads();

    const int rs1 = rs[1];
    const int rs2 = rs[2];
    const int E   = (N + 1023) / 1024;
    const int base = tid * E;

    int cnt = 0, c1 = 0, c2 = 0;
    for (int k = 0; k < E; ++k) {
        const int j = base + k;
        if (j < N && score[j] > THRV) {
            ++cnt;
            if (j < rs1) ++c1;
            if (j < rs2) ++c2;
        }
    }
    atomicAdd(&s_c1, c1);
    atomicAdd(&s_c2, c2);
    s_scan[tid] = cnt;
    __syncthreads();

    // Hillis-Steele inclusive scan over 1024 per-thread counts
    for (int off = 1; off < 1024; off <<= 1) {
        const int v   = s_scan[tid];
        const int add = (tid >= off) ? s_scan[tid - off] : 0;
        __syncthreads();
        s_scan[tid] = v + add;
        __syncthreads();
    }
    int pos = s_scan[tid] - cnt;   // exclusive prefix

    for (int k = 0; k < E; ++k) {
        const int j = base + k;
        if (j < N && score[j] > THRV) {
            if (pos < M) out[off_sel + pos] = (float)j;
            ++pos;
        }
    }

    if (tid == 0) {
        out[0] = (float)rs[0];
        out[1] = (float)rs1;
        out[2] = (float)rs2;
        out[3] = 0.0f;
        out[4] = (float)s_c1;
        out[5] = (float)s_c2;
    }
}

// ---------------------------------------------------------------------------
extern "C" void kernel_launch(void* const* d_in, const int* in_sizes, int n_in,
                              void* d_out, int out_size, void* d_ws, size_t ws_size,
                              hipStream_t stream)
{
    const float* score  = (const float*)d_in[0];
    const float* coords = (const float*)d_in[1];
    const int*   rs     = (const int*)d_in[2];

    const int  N  = in_sizes[0];          // 8192
    const long NK = (long)N * KNN;

    float* out = (float*)d_out;
    const int  M        = out_size - 6 - (int)(3 * NK);
    const long off_nidx = 6;
    const long off_dist = 6 + NK;
    const long off_sel  = 6 + 2 * NK;
    const long off_w    = off_sel + M;

    compact_kernel<<<1, 1024, 0, stream>>>(score, rs, out, N, (int)off_sel, M);

    knn_down_kernel<<<N / TS, BLK, 0, stream>>>(
        score, coords, rs,
        out + off_nidx, out + off_dist, out + off_w, N);
}